// EpisodicMetaController_62912680951998
// MI455X (gfx1250) — compile-verified
//
#include <hip/hip_runtime.h>
#include <hip/hip_bf16.h>
#include <math.h>

// Guard only in the DEVICE pass: __has_builtin() for amdgcn builtins returns 0
// in the host pass even when the builtin is perfectly usable in device code.
#if defined(__HIP_DEVICE_COMPILE__)
#if !__has_builtin(__builtin_amdgcn_wmma_f32_16x16x4_f32)
#error "gfx1250 f32 WMMA builtin __builtin_amdgcn_wmma_f32_16x16x4_f32 not available"
#endif
#endif

typedef __attribute__((ext_vector_type(2))) float v2f;
typedef __attribute__((ext_vector_type(8))) float v8f;

#define BB   2048
#define HH   256
#define KS   32
#define UU   128
#define OBSD 128
#define SSUMD 128
#define PROJD 128
#define XIN  256
#define N3H  768
#define CTXD 512

__device__ __forceinline__ float sigmoidf_(float x) { return 1.0f / (1.0f + expf(-x)); }

__device__ __forceinline__ float wave_sum(float v) {
#pragma unroll
    for (int o = 16; o > 0; o >>= 1) v += __shfl_xor(v, o, 32);
    return v;
}

__device__ __forceinline__ float block_sum256(float v, float* shred, int wave, int lane) {
    v = wave_sum(v);
    __syncthreads();                 // protect shred from previous use
    if (lane == 0) shred[wave] = v;
    __syncthreads();
    float t = 0.f;
#pragma unroll
    for (int i = 0; i < 8; ++i) t += shred[i];
    return t;
}

// ---------------------------------------------------------------------------
// Weight transpose: dst[n*K + k] = src[k*N + n]   (weights are tiny, L2-hot)
// ---------------------------------------------------------------------------
__global__ void transpose_kernel(const float* __restrict__ src, float* __restrict__ dst,
                                 int K, int N) {
    int idx = blockIdx.x * 256 + threadIdx.x;
    if (idx < K * N) {
        int k = idx / N;
        int n = idx - k * N;
        dst[n * K + k] = src[idx];
    }
}

// ---------------------------------------------------------------------------
// Generic fp32 WMMA GEMM:  C[M x ldc](+coff) = epi(A[M x KDIM] @ B[KDIM x NDIM] + bias)
// Bt is pre-transposed: Bt[NDIM x KDIM].
// Block = 256 threads (8 waves), one 16-row M tile per block, A panel in LDS.
// Wave w handles N tiles w, w+8, ...
// EPI: 0=none, 1=+bias, 2=tanh(+bias), 3=u_int head (u_prop + u_int blend)
// ---------------------------------------------------------------------------
template <int KDIM, int NDIM, int EPI>
__global__ __launch_bounds__(256) void gemm_wmma_f32(
    const float* __restrict__ A,
    const float* __restrict__ Bt,
    const float* __restrict__ bias,
    float* __restrict__ C, int ldc, int coff,
    const float* __restrict__ rowScale,   // beta_eff per row   (EPI==3)
    const float* __restrict__ uPrev,      // u_int_prev [M x NDIM] (EPI==3)
    float* __restrict__ uInt)             // u_int out  [M x NDIM] (EPI==3)
{
    __shared__ float As[16 * KDIM];

    const int mtile = blockIdx.x;                  // 16 rows
    const int tid   = threadIdx.x;
    const int wave  = tid >> 5;
    const int lane  = tid & 31;
    const int l16   = lane & 15;                   // A: row m   |  B/C: col n
    const int half  = lane >> 4;

    // Cooperative coalesced A-panel load (float4)
    {
        const float4* A4  = (const float4*)(A + (size_t)mtile * 16 * KDIM);
        float4*       As4 = (float4*)As;
#pragma unroll
        for (int i = tid; i < 16 * KDIM / 4; i += 256) As4[i] = A4[i];
    }
    __syncthreads();

    for (int nt = wave; nt < NDIM / 16; nt += 8) {
        const int n = nt * 16 + l16;
        const float* btrow = Bt + (size_t)n * KDIM;
        v8f acc = {};
#pragma unroll 8
        for (int k0 = 0; k0 < KDIM; k0 += 4) {
            v2f a  = *(const v2f*)(&As[l16 * KDIM + k0 + 2 * half]);   // A[m][k..k+1] / [k+2..k+3]
            v2f bv = *(const v2f*)(&btrow[k0 + 2 * half]);             // B[k..][n] via Bt
            acc = __builtin_amdgcn_wmma_f32_16x16x4_f32(
                false, a, false, bv, (short)0, acc, false, false);
        }
        const float bsv = (EPI == 0) ? 0.f : bias[n];
#pragma unroll
        for (int j = 0; j < 8; ++j) {
            int   row = mtile * 16 + j + half * 8;
            float v   = acc[j] + bsv;
            if (EPI == 2) v = tanhf(v);
            C[(size_t)row * ldc + coff + n] = v;
            if (EPI == 3) {
                float be = rowScale[row];
                float up = uPrev[(size_t)row * NDIM + n];
                uInt[(size_t)row * NDIM + n] = be * up + (1.f - be) * v;
            }
        }
    }
}

// ---------------------------------------------------------------------------
// GRU elementwise: h_t from gi (with bias already), gh (no bias), bhn, h_prev
// ---------------------------------------------------------------------------
__global__ __launch_bounds__(256) void gru_kernel(
    const float* __restrict__ gi, const float* __restrict__ gh,
    const float* __restrict__ bhn, const float* __restrict__ h_prev,
    float* __restrict__ h_t)
{
    int idx = blockIdx.x * 256 + threadIdx.x;      // 0 .. B*H-1
    int b   = idx >> 8;
    int h   = idx & 255;
    const float* gib = gi + (size_t)b * N3H;
    const float* ghb = gh + (size_t)b * N3H;
    float ir = gib[h], iz = gib[h + HH], inn = gib[h + 2 * HH];
    float hr = ghb[h], hz = ghb[h + HH], hn  = ghb[h + 2 * HH];
    float r = sigmoidf_(ir + hr);
    float z = sigmoidf_(iz + hz);
    float n = tanhf(inn + r * (hn + bhn[h]));
    h_t[idx] = (1.f - z) * n + z * h_prev[idx];
}

// ---------------------------------------------------------------------------
// Episodic memory kernel: one block (256 thr) per batch row.
// ---------------------------------------------------------------------------
__global__ __launch_bounds__(256) void episodic_kernel(
    const float* __restrict__ h_t,
    const float* __restrict__ keys, const float* __restrict__ vals,
    const float* __restrict__ age,  const float* __restrict__ strength,
    const float* __restrict__ W_gate, const float* __restrict__ b_gate,
    const float* __restrict__ W_beta, const float* __restrict__ b_beta,
    const float* __restrict__ W_sw,   const float* __restrict__ b_sw,
    const int*   __restrict__ step_in_macro,
    float* __restrict__ keys_new, float* __restrict__ vals_new,
    float* __restrict__ age_new,  float* __restrict__ str_new,
    float* __restrict__ beta_out, float* __restrict__ sw_out,
    int*   __restrict__ switch_out,
    float* __restrict__ ctx, float* __restrict__ beta_eff)
{
    __shared__ float sh_qn[HH];
    __shared__ float sh_wread[KS];
    __shared__ float sh_rate[KS];
    __shared__ float shred[8];
    __shared__ float sh_sim[KS];

    const int b    = blockIdx.x;
    const int tid  = threadIdx.x;
    const int wave = tid >> 5;
    const int lane = tid & 31;

    const float h = h_t[(size_t)b * HH + tid];

    // ||h||, qn
    float nrm2 = block_sum256(h * h, shred, wave, lane);
    float inv  = 1.f / (sqrtf(nrm2) + 1e-6f);
    float qn   = h * inv;
    sh_qn[tid] = qn;

    // w_str = sigmoid(h . W_gate + b_gate)   (block_sum256 barriers cover sh_qn)
    float gdot = block_sum256(h * W_gate[tid], shred, wave, lane);
    const float ws_val = sigmoidf_(gdot + b_gate[0]);

    // slot similarities: wave w -> slots 4w..4w+3, 32 lanes x 8 elems each
#pragma unroll
    for (int i = 0; i < 4; ++i) {
        int s = wave * 4 + i;
        const float* krow = keys + ((size_t)b * KS + s) * HH;
        float dp = 0.f, kk = 0.f;
#pragma unroll
        for (int j = lane; j < HH; j += 32) {
            float kv = krow[j];
            dp += kv * sh_qn[j];
            kk += kv * kv;
        }
        dp = wave_sum(dp);
        kk = wave_sum(kk);
        if (lane == 0) sh_sim[s] = dp / (sqrtf(kk) + 1e-6f);
    }
    __syncthreads();

    // wave 0: per-slot softmax (read) + hard write slot + age/strength update
    if (wave == 0) {
        int   s   = lane;
        float sim = sh_sim[s];
        float ag  = age[(size_t)b * KS + s];
        float st  = strength[(size_t)b * KS + s];

        // read softmax
        float logit = sim + 0.5f * logf(fminf(fmaxf(st, 0.001f), 1.0f)) - 0.02f * ag;
        float mx = logit;
#pragma unroll
        for (int o = 16; o > 0; o >>= 1) mx = fmaxf(mx, __shfl_xor(mx, o, 32));
        float e   = expf(logit - mx);
        float sum = wave_sum(e);
        sh_wread[s] = e / sum;

        // write logits; forward value of write_w == one_hot(argmax)
        float wl  = sim * 50.0f + 0.05f * log1pf(ag) - 0.5f * st;
        float m   = wl;
        int   idx = s;
#pragma unroll
        for (int o = 16; o > 0; o >>= 1) {
            float om = __shfl_xor(m, o, 32);
            int   oi = __shfl_xor(idx, o, 32);
            if (om > m || (om == m && oi < idx)) { m = om; idx = oi; }
        }
        float ww = (s == idx) ? 1.f : 0.f;
        sh_rate[s] = ww * ws_val * 0.5f;   // EPI_WRITE_ALPHA = 0.5

        age_new[(size_t)b * KS + s] = (ag + 1.f) * (1.f - ww);
        float sn = st * 0.995f;            // EPI_STRENGTH_DECAY
        sn = sn + ww * ws_val * (1.f - sn);
        str_new[(size_t)b * KS + s] = fminf(fmaxf(sn, 0.f), 1.f);
    }
    __syncthreads();

    // fused sweep: read_out + keys_new/vals_new (single coalesced RMW pass)
    float ro = 0.f;
    const float qv = sh_qn[tid];
#pragma unroll 4
    for (int s = 0; s < KS; ++s) {
        size_t off = ((size_t)b * KS + s) * HH + tid;
        float kv = keys[off];
        float vv = vals[off];
        ro += sh_wread[s] * vv;
        float rt = sh_rate[s];
        keys_new[off] = (1.f - rt) * kv + rt * qv;
        vals_new[off] = (1.f - rt) * vv + rt * h;
    }

    // ctx = [h_t, read_out]
    ctx[(size_t)b * CTXD + tid]      = h;
    ctx[(size_t)b * CTXD + HH + tid] = ro;

    // beta / sw heads over ctx (512-dot each)
    float bd = block_sum256(h * W_beta[tid] + ro * W_beta[HH + tid], shred, wave, lane);
    float sd = block_sum256(h * W_sw[tid]   + ro * W_sw[HH + tid],   shred, wave, lane);
    if (tid == 0) {
        float beta = 0.05f + (0.995f - 0.05f) * sigmoidf_(bd + b_beta[0]);
        float swp  = sigmoidf_(sd + b_sw[0]);
        float swf  = (swp > 0.5f) ? 1.f : 0.f;          // SWITCH_TAU
        if (step_in_macro[b] == 0) swf = 1.f;
        beta_out[b]   = beta;
        sw_out[b]     = swp;
        switch_out[b] = (int)swf;
        beta_eff[b]   = beta * (1.f - swf);
    }
}

// ---------------------------------------------------------------------------
extern "C" void kernel_launch(void* const* d_in, const int* in_sizes, int n_in,
                              void* d_out, int out_size, void* d_ws, size_t ws_size,
                              hipStream_t stream) {
    const float* h_prev   = (const float*)d_in[0];
    const float* keys     = (const float*)d_in[1];
    const float* vals     = (const float*)d_in[2];
    const float* age      = (const float*)d_in[3];
    const float* strength = (const float*)d_in[4];
    const float* obs      = (const float*)d_in[5];
    const float* ssum     = (const float*)d_in[6];
    const float* u_prev   = (const float*)d_in[7];
    const int*   step     = (const int*)  d_in[8];
    const float* W_obs    = (const float*)d_in[9];
    const float* b_obs    = (const float*)d_in[10];
    const float* W_ssum   = (const float*)d_in[11];
    const float* b_ssum   = (const float*)d_in[12];
    const float* Wi       = (const float*)d_in[13];
    const float* bi       = (const float*)d_in[14];
    const float* Wh       = (const float*)d_in[15];
    const float* bhn      = (const float*)d_in[16];
    const float* W_gate   = (const float*)d_in[17];
    const float* b_gate   = (const float*)d_in[18];
    const float* W_mean   = (const float*)d_in[19];
    const float* b_mean   = (const float*)d_in[20];
    /* d_in[21] = u_logstd, unused by reference outputs */
    const float* W_beta   = (const float*)d_in[22];
    const float* b_beta   = (const float*)d_in[23];
    const float* W_sw     = (const float*)d_in[24];
    const float* b_sw     = (const float*)d_in[25];

    // ---- output layout (flat, reference return order) ----
    float* out      = (float*)d_out;
    float* o_ht     = out;                        // B*H
    float* o_keys   = o_ht   + (size_t)BB * HH;   // B*K*H
    float* o_vals   = o_keys + (size_t)BB * KS * HH;
    float* o_age    = o_vals + (size_t)BB * KS * HH;
    float* o_str    = o_age  + (size_t)BB * KS;
    float* o_uprop  = o_str  + (size_t)BB * KS;   // B*U
    int*   o_switch = (int*)(o_uprop + (size_t)BB * UU);
    float* o_uint   = (float*)o_switch + BB;      // B*U
    float* o_beta   = o_uint + (size_t)BB * UU;
    float* o_sw     = o_beta + BB;
    (void)o_sw; (void)out_size;

    // ---- workspace carve ----
    float* ws      = (float*)d_ws;
    float* x_in    = ws;                   ws += (size_t)BB * XIN;    // 2 MB
    float* gi      = ws;                   ws += (size_t)BB * N3H;    // 6 MB
    float* gh      = ws;                   ws += (size_t)BB * N3H;    // 6 MB
    float* ctx     = ws;                   ws += (size_t)BB * CTXD;   // 4 MB
    float* beta_eff= ws;                   ws += BB;
    float* Wt_obs  = ws;                   ws += OBSD * PROJD;
    float* Wt_ssum = ws;                   ws += SSUMD * PROJD;
    float* Wt_i    = ws;                   ws += XIN * N3H;
    float* Wt_h    = ws;                   ws += HH * N3H;
    float* Wt_mean = ws;                   ws += CTXD * UU;
    (void)ws_size;

    // 1) transpose weights for WMMA B-operand (contiguous float2 per lane)
    transpose_kernel<<<(OBSD * PROJD + 255) / 256, 256, 0, stream>>>(W_obs,  Wt_obs,  OBSD, PROJD);
    transpose_kernel<<<(SSUMD * PROJD + 255) / 256, 256, 0, stream>>>(W_ssum, Wt_ssum, SSUMD, PROJD);
    transpose_kernel<<<(XIN * N3H + 255) / 256, 256, 0, stream>>>(Wi, Wt_i, XIN, N3H);
    transpose_kernel<<<(HH * N3H + 255) / 256, 256, 0, stream>>>(Wh, Wt_h, HH, N3H);
    transpose_kernel<<<(CTXD * UU + 255) / 256, 256, 0, stream>>>(W_mean, Wt_mean, CTXD, UU);

    const int mblocks = BB / 16;   // 128

    // 2) x_in = [tanh(obs@W_obs+b) | tanh(ssum@W_ssum+b)]
    gemm_wmma_f32<OBSD, PROJD, 2><<<mblocks, 256, 0, stream>>>(
        obs, Wt_obs, b_obs, x_in, XIN, 0, nullptr, nullptr, nullptr);
    gemm_wmma_f32<SSUMD, PROJD, 2><<<mblocks, 256, 0, stream>>>(
        ssum, Wt_ssum, b_ssum, x_in, XIN, PROJD, nullptr, nullptr, nullptr);

    // 3) gi = x_in@Wi + bi ; gh = h_prev@Wh
    gemm_wmma_f32<XIN, N3H, 1><<<mblocks, 256, 0, stream>>>(
        x_in, Wt_i, bi, gi, N3H, 0, nullptr, nullptr, nullptr);
    gemm_wmma_f32<HH, N3H, 0><<<mblocks, 256, 0, stream>>>(
        h_prev, Wt_h, nullptr, gh, N3H, 0, nullptr, nullptr, nullptr);

    // 4) GRU -> h_t
    gru_kernel<<<(BB * HH) / 256, 256, 0, stream>>>(gi, gh, bhn, h_prev, o_ht);

    // 5) episodic memory: sims, softmax, fused read_out + keys/vals update, heads
    episodic_kernel<<<BB, 256, 0, stream>>>(
        o_ht, keys, vals, age, strength,
        W_gate, b_gate, W_beta, b_beta, W_sw, b_sw, step,
        o_keys, o_vals, o_age, o_str,
        o_beta, o_sw, o_switch, ctx, beta_eff);

    // 6) mean head: u_prop = ctx@W_mean + b_mean ; u_int = blend(beta_eff, u_prev, u_prop)
    gemm_wmma_f32<CTXD, UU, 3><<<mblocks, 256, 0, stream>>>(
        ctx, Wt_mean, b_mean, o_uprop, UU, 0, beta_eff, u_prev, o_uint);
}